// GAT_MNIST_82197084111194
// MI455X (gfx1250) — compile-verified
//
#include <hip/hip_runtime.h>
#include <math.h>

// ---------------- problem constants ----------------
static constexpr int Nn   = 50000;     // nodes
static constexpr int Ee   = 800000;    // edges before self loops
static constexpr int Et   = Ee + Nn;   // edges incl self loops
static constexpr int Gg   = 500;       // graphs
static constexpr int Hh   = 4;         // heads
static constexpr int HCc  = 128;       // H*C
static constexpr int MLPd = 128;
static constexpr int NCc  = 10;
static constexpr int Ff   = 3;

typedef float v2f __attribute__((ext_vector_type(2)));
typedef float v8f __attribute__((ext_vector_type(8)));

// ---------------- helpers ----------------
__device__ inline void atomicMaxFloat(float* addr, float val) {
    unsigned int* ua = (unsigned int*)addr;
    unsigned int old = __float_as_uint(*addr);
    while (__uint_as_float(old) < val) {
        unsigned int assumed = old;
        old = atomicCAS(ua, assumed, __float_as_uint(val));
        if (old == assumed) break;
    }
}

__global__ void fill_f32(float* __restrict__ p, int n, float v) {
    int i = blockIdx.x * blockDim.x + threadIdx.x;
    if (i < n) p[i] = v;
}

// ---------------- fast f32 WMMA GEMM (aligned dims) ----------------
// Requires M%16==0, K%4==0, Ncols%64==0.  One wave per 16x64 output strip:
// 4 accumulators share one A fragment -> 4 v_wmma per K-step, branch-free
// inner loop (EXEC stays all-ones; unconditional vector loads).
__global__ void gemm_wmma_f32_fast(const float* __restrict__ A, const float* __restrict__ B,
                                   float* __restrict__ C, int M, int K, int Ncols,
                                   int colGroups) {
    int gid = blockIdx.x;
    int tm = gid / colGroups;
    int cg = gid - tm * colGroups;
    int lane = threadIdx.x;          // 0..31
    int half = lane >> 4;
    int l16  = lane & 15;
    const float* Arow = A + (size_t)(tm * 16 + l16) * K;
    int n0 = cg * 64 + l16;
    v8f acc0 = {}, acc1 = {}, acc2 = {}, acc3 = {};
#pragma unroll 2
    for (int k0 = 0; k0 < K; k0 += 4) {
        int kb = k0 + 2 * half;
        v2f a = *(const v2f*)(Arow + kb);            // contiguous K pair
        const float* Bp = B + (size_t)kb * Ncols + n0;
        v2f b0, b1, b2, b3;
        b0[0] = Bp[0];  b0[1] = Bp[Ncols + 0];
        b1[0] = Bp[16]; b1[1] = Bp[Ncols + 16];
        b2[0] = Bp[32]; b2[1] = Bp[Ncols + 32];
        b3[0] = Bp[48]; b3[1] = Bp[Ncols + 48];
        acc0 = __builtin_amdgcn_wmma_f32_16x16x4_f32(false, a, false, b0, (short)0, acc0, false, false);
        acc1 = __builtin_amdgcn_wmma_f32_16x16x4_f32(false, a, false, b1, (short)0, acc1, false, false);
        acc2 = __builtin_amdgcn_wmma_f32_16x16x4_f32(false, a, false, b2, (short)0, acc2, false, false);
        acc3 = __builtin_amdgcn_wmma_f32_16x16x4_f32(false, a, false, b3, (short)0, acc3, false, false);
    }
#pragma unroll
    for (int r = 0; r < 8; ++r) {
        int row = tm * 16 + r + 8 * half;
        float* Cp = C + (size_t)row * Ncols + n0;
        Cp[0]  = acc0[r];
        Cp[16] = acc1[r];
        Cp[32] = acc2[r];
        Cp[48] = acc3[r];
    }
}

// ---------------- generic f32 WMMA GEMM (guarded, branch-free) ----------------
// C[M,Ncols] = A[M,K] @ B[K,Ncols] (+ bias).  One wave per 16x16 tile.
// Guards are done by clamping addresses (always-valid unconditional load)
// followed by a select -> v_cndmask, no EXEC branching in the inner loop.
__global__ void gemm_wmma_f32(const float* __restrict__ A, const float* __restrict__ B,
                              const float* __restrict__ bias, float* __restrict__ C,
                              int M, int K, int Ncols, int tilesN) {
    int tile = blockIdx.x;
    int tm = tile / tilesN;
    int tn = tile - tm * tilesN;
    int lane = threadIdx.x;
    int half = lane >> 4;
    int l16  = lane & 15;
    int m = tm * 16 + l16;
    int n = tn * 16 + l16;
    int mc = min(m, M - 1);
    int nc = min(n, Ncols - 1);
    v8f acc = {};
#pragma unroll 4
    for (int k0 = 0; k0 < K; k0 += 4) {
        v2f a, b;
#pragma unroll
        for (int v = 0; v < 2; ++v) {
            int kk = k0 + v + 2 * half;
            int kc = min(kk, K - 1);
            float av = A[(size_t)mc * K + kc];       // always in-bounds
            float bv = B[(size_t)kc * Ncols + nc];   // always in-bounds
            a[v] = (m < M && kk < K) ? av : 0.0f;    // v_cndmask, no branch
            b[v] = (kk < K && n < Ncols) ? bv : 0.0f;
        }
        acc = __builtin_amdgcn_wmma_f32_16x16x4_f32(
                  false, a, false, b, (short)0, acc, false, false);
    }
#pragma unroll
    for (int r = 0; r < 8; ++r) {
        int row = tm * 16 + r + 8 * half;
        if (row < M && n < Ncols) {
            float v = acc[r];
            if (bias) v += bias[n];
            C[(size_t)row * Ncols + n] = v;
        }
    }
}

// ---------------- attention logits per node ----------------
__global__ void node_logits(const float* __restrict__ h, const float* __restrict__ a_s,
                            const float* __restrict__ a_d,
                            float* __restrict__ es, float* __restrict__ ed) {
    int idx = blockIdx.x * blockDim.x + threadIdx.x;
    if (idx >= Nn * Hh) return;
    int n = idx >> 2, hd = idx & 3;
    const float4* hp = (const float4*)(h + n * HCc + hd * 32);
    const float4* sp = (const float4*)(a_s + hd * 32);
    const float4* dp = (const float4*)(a_d + hd * 32);
    float s = 0.f, t = 0.f;
#pragma unroll
    for (int i = 0; i < 8; ++i) {
        float4 v = hp[i], a = sp[i], b = dp[i];
        s += v.x * a.x + v.y * a.y + v.z * a.z + v.w * a.w;
        t += v.x * b.x + v.y * b.y + v.z * b.z + v.w * b.w;
    }
    es[idx] = s;
    ed[idx] = t;
}

// ---------------- edge pass 1: leaky-relu logit + segment max ----------------
__global__ void edge_logit_max(const int* __restrict__ ei, const float* __restrict__ es,
                               const float* __restrict__ ed, float* __restrict__ eb,
                               float* __restrict__ maxb) {
    int idx = blockIdx.x * blockDim.x + threadIdx.x;
    if (idx >= Et * Hh) return;
    int e = idx >> 2, hd = idx & 3;
    int s = (e < Ee) ? ei[e] : (e - Ee);
    int d = (e < Ee) ? ei[Ee + e] : (e - Ee);
    float v = es[s * Hh + hd] + ed[d * Hh + hd];
    v = (v >= 0.f) ? v : 0.2f * v;                  // leaky_relu 0.2
    eb[idx] = v;
    atomicMaxFloat(&maxb[d * Hh + hd], v);
}

// ---------------- edge pass 2: exp + segment sum ----------------
__global__ void edge_exp_sum(const int* __restrict__ ei, float* __restrict__ eb,
                             const float* __restrict__ maxb, float* __restrict__ sumb) {
    int idx = blockIdx.x * blockDim.x + threadIdx.x;
    if (idx >= Et * Hh) return;
    int e = idx >> 2, hd = idx & 3;
    int d = (e < Ee) ? ei[Ee + e] : (e - Ee);
    float m = maxb[d * Hh + hd];
    if (m < -3.0e38f) m = 0.f;                      // isfinite fixup
    float p = expf(eb[idx] - m);
    eb[idx] = p;
    atomicAdd(&sumb[d * Hh + hd], p);
}

// ---------------- edge pass 3: weighted aggregation (wave per edge) ----------
__global__ void edge_aggregate(const int* __restrict__ ei, const float* __restrict__ hfeat,
                               const float* __restrict__ eb, const float* __restrict__ sumb,
                               float* __restrict__ obuf) {
    int e = blockIdx.x * (blockDim.x >> 5) + (threadIdx.x >> 5);
    if (e >= Et) return;
    int lane = threadIdx.x & 31;
    int s = (e < Ee) ? ei[e] : (e - Ee);
    int d = (e < Ee) ? ei[Ee + e] : (e - Ee);
    int hd = lane >> 3;                              // 4 channels/lane -> head = lane/8
    float alpha = eb[e * Hh + hd] / (sumb[d * Hh + hd] + 1e-16f);
    float4 hv = ((const float4*)(hfeat + (size_t)s * HCc))[lane];
    float* ob = obuf + (size_t)d * HCc + lane * 4;
    atomicAdd(ob + 0, hv.x * alpha);
    atomicAdd(ob + 1, hv.y * alpha);
    atomicAdd(ob + 2, hv.z * alpha);
    atomicAdd(ob + 3, hv.w * alpha);
}

// ---------------- batch-norm (two pass) ----------------
__global__ void bn_stats(const float* __restrict__ x, float* __restrict__ stats,
                         int M, int Cch) {
    int c = threadIdx.x;                             // blockDim.x == Cch
    float s = 0.f, q = 0.f;
    for (int r = blockIdx.x; r < M; r += gridDim.x) {
        float v = x[(size_t)r * Cch + c];
        s += v;
        q += v * v;
    }
    atomicAdd(&stats[c], s);
    atomicAdd(&stats[Cch + c], q);
}

__global__ void bn_apply(float* __restrict__ x, const float* __restrict__ stats,
                         const float* __restrict__ g, const float* __restrict__ be,
                         int M, int Cch, int relu) {
    int idx = blockIdx.x * blockDim.x + threadIdx.x;
    if (idx >= M * Cch) return;
    int c = idx % Cch;
    float inv = 1.0f / (float)M;
    float mu = stats[c] * inv;
    float var = stats[Cch + c] * inv - mu * mu;
    float v = (x[idx] - mu) * rsqrtf(var + 1e-5f) * g[c] + be[c];
    if (relu) v = fmaxf(v, 0.f);
    x[idx] = v;
}

// ---------------- pooling ----------------
__global__ void pool_scatter(const float* __restrict__ x, const int* __restrict__ bidx,
                             float* __restrict__ gmax, float* __restrict__ gsum,
                             float* __restrict__ cnt) {
    int idx = blockIdx.x * blockDim.x + threadIdx.x;
    if (idx >= Nn * HCc) return;
    int n = idx >> 7, c = idx & 127;
    int g = bidx[n];
    float v = x[idx];
    atomicMaxFloat(&gmax[g * HCc + c], v);
    atomicAdd(&gsum[g * HCc + c], v);
    if (c == 0) atomicAdd(&cnt[g], 1.0f);
}

__global__ void pool_finish(const float* __restrict__ gmax, const float* __restrict__ gsum,
                            const float* __restrict__ cnt, float* __restrict__ hcat) {
    int idx = blockIdx.x * blockDim.x + threadIdx.x;
    if (idx >= Gg * HCc) return;
    int g = idx >> 7, c = idx & 127;
    float mx = gmax[idx];
    if (mx < -3.0e38f) mx = 0.f;                     // isfinite fixup
    hcat[g * 2 * HCc + c] = mx;
    float ct = fmaxf(cnt[g], 1.0f);
    hcat[g * 2 * HCc + HCc + c] = gsum[idx] / ct;
}

// ---------------- host-side orchestration ----------------
extern "C" void kernel_launch(void* const* d_in, const int* in_sizes, int n_in,
                              void* d_out, int out_size, void* d_ws, size_t ws_size,
                              hipStream_t stream) {
    (void)in_sizes; (void)n_in; (void)out_size; (void)ws_size;
    const float* x    = (const float*)d_in[0];
    const int*   ei   = (const int*)d_in[1];
    const int*   bidx = (const int*)d_in[2];
    const float* W[3]  = {(const float*)d_in[3],  (const float*)d_in[9],  (const float*)d_in[15]};
    const float* As[3] = {(const float*)d_in[4],  (const float*)d_in[10], (const float*)d_in[16]};
    const float* Ad[3] = {(const float*)d_in[5],  (const float*)d_in[11], (const float*)d_in[17]};
    const float* Gm[3] = {(const float*)d_in[7],  (const float*)d_in[13], (const float*)d_in[19]};
    const float* Be[3] = {(const float*)d_in[8],  (const float*)d_in[14], (const float*)d_in[20]};
    // layer biases b1..b3, bm1, bm2 cancel exactly under BN mean-subtraction.
    const float* Wm1 = (const float*)d_in[21];
    const float* gm1 = (const float*)d_in[23];
    const float* bem1= (const float*)d_in[24];
    const float* Wm2 = (const float*)d_in[25];
    const float* gm2 = (const float*)d_in[27];
    const float* bem2= (const float*)d_in[28];
    const float* Wm3 = (const float*)d_in[29];
    const float* bm3 = (const float*)d_in[30];
    float* out = (float*)d_out;

    // workspace layout (floats)
    float* ws    = (float*)d_ws;
    float* hbuf  = ws;                 ws += (size_t)Nn * HCc;
    float* o1    = ws;                 ws += (size_t)Nn * HCc;
    float* o2    = ws;                 ws += (size_t)Nn * HCc;
    float* es    = ws;                 ws += (size_t)Nn * Hh;
    float* ed    = ws;                 ws += (size_t)Nn * Hh;
    float* maxb  = ws;                 ws += (size_t)Nn * Hh;
    float* sumb  = ws;                 ws += (size_t)Nn * Hh;
    float* pedge = ws;                 ws += (size_t)Et * Hh;
    float* stats = ws;                 ws += 256;
    float* gmax  = ws;                 ws += (size_t)Gg * HCc;
    float* gsum  = ws;                 ws += (size_t)Gg * HCc;
    float* cnt   = ws;                 ws += Gg;
    float* hcat  = ws;                 ws += (size_t)Gg * 2 * HCc;
    float* m1    = ws;                 ws += (size_t)Gg * MLPd;
    float* m2    = ws;                 ws += (size_t)Gg * MLPd;

    const int TB = 256;
    auto blocks = [](long n, int tb) { return (int)((n + tb - 1) / tb); };

    auto run_layer = [&](const float* xin, int K, int l, float* ob) {
        if (K % 4 == 0) {
            // aligned fast path: 16x64 strip per wave, 4 wmma per K-step
            int colGroups = HCc / 64;                         // 2
            gemm_wmma_f32_fast<<<(Nn / 16) * colGroups, 32, 0, stream>>>(
                xin, W[l], hbuf, Nn, K, HCc, colGroups);
        } else {
            int tilesM = (Nn + 15) / 16, tilesN = HCc / 16;
            gemm_wmma_f32<<<tilesM * tilesN, 32, 0, stream>>>(xin, W[l], nullptr, hbuf,
                                                              Nn, K, HCc, tilesN);
        }
        node_logits<<<blocks((long)Nn * Hh, TB), TB, 0, stream>>>(hbuf, As[l], Ad[l], es, ed);
        fill_f32<<<blocks((long)Nn * Hh, TB), TB, 0, stream>>>(maxb, Nn * Hh, -INFINITY);
        fill_f32<<<blocks((long)Nn * Hh, TB), TB, 0, stream>>>(sumb, Nn * Hh, 0.f);
        fill_f32<<<blocks((long)Nn * HCc, TB), TB, 0, stream>>>(ob, Nn * HCc, 0.f);
        edge_logit_max<<<blocks((long)Et * Hh, TB), TB, 0, stream>>>(ei, es, ed, pedge, maxb);
        edge_exp_sum<<<blocks((long)Et * Hh, TB), TB, 0, stream>>>(ei, pedge, maxb, sumb);
        edge_aggregate<<<blocks((long)Et * 32, TB), TB, 0, stream>>>(ei, hbuf, pedge, sumb, ob);
        fill_f32<<<1, 256, 0, stream>>>(stats, 256, 0.f);
        bn_stats<<<256, HCc, 0, stream>>>(ob, stats, Nn, HCc);
        bn_apply<<<blocks((long)Nn * HCc, TB), TB, 0, stream>>>(ob, stats, Gm[l], Be[l],
                                                                Nn, HCc, 0);
    };

    run_layer(x,  Ff,  0, o1);   // layer 1: [N,3] -> o1   (generic, K=3)
    run_layer(o1, HCc, 1, o2);   // layer 2: o1 -> o2      (fast path)
    run_layer(o2, HCc, 2, o1);   // layer 3: o2 -> o1      (fast path)

    // pooling -> hcat [G, 256]
    fill_f32<<<blocks((long)Gg * HCc, TB), TB, 0, stream>>>(gmax, Gg * HCc, -INFINITY);
    fill_f32<<<blocks((long)Gg * HCc, TB), TB, 0, stream>>>(gsum, Gg * HCc, 0.f);
    fill_f32<<<blocks(Gg, TB), TB, 0, stream>>>(cnt, Gg, 0.f);
    pool_scatter<<<blocks((long)Nn * HCc, TB), TB, 0, stream>>>(o1, bidx, gmax, gsum, cnt);
    pool_finish<<<blocks((long)Gg * HCc, TB), TB, 0, stream>>>(gmax, gsum, cnt, hcat);

    // MLP head (small, guarded generic kernel)
    int tmG = (Gg + 15) / 16;                          // 32 row tiles
    gemm_wmma_f32<<<tmG * (MLPd / 16), 32, 0, stream>>>(hcat, Wm1, nullptr, m1,
                                                        Gg, 2 * HCc, MLPd, MLPd / 16);
    fill_f32<<<1, 256, 0, stream>>>(stats, 256, 0.f);
    bn_stats<<<256, MLPd, 0, stream>>>(m1, stats, Gg, MLPd);
    bn_apply<<<blocks((long)Gg * MLPd, TB), TB, 0, stream>>>(m1, stats, gm1, bem1, Gg, MLPd, 1);

    gemm_wmma_f32<<<tmG * (MLPd / 16), 32, 0, stream>>>(m1, Wm2, nullptr, m2,
                                                        Gg, MLPd, MLPd, MLPd / 16);
    fill_f32<<<1, 256, 0, stream>>>(stats, 256, 0.f);
    bn_stats<<<256, MLPd, 0, stream>>>(m2, stats, Gg, MLPd);
    bn_apply<<<blocks((long)Gg * MLPd, TB), TB, 0, stream>>>(m2, stats, gm2, bem2, Gg, MLPd, 1);

    gemm_wmma_f32<<<tmG * 1, 32, 0, stream>>>(m2, Wm3, bm3, out, Gg, MLPd, NCc, 1);
}